// GraphEncoder_42966852829219
// MI455X (gfx1250) — compile-verified
//
#include <hip/hip_runtime.h>
#include <hip/hip_bf16.h>

typedef __attribute__((ext_vector_type(2))) float v2f;
typedef __attribute__((ext_vector_type(8))) float v8f;

#define DOUT 128

// ---------------------------------------------------------------------------
// GEMM: Y[M x 128] = act(X)[M x K] @ W[K x 128] + bias
// One wave32 computes a 16-row x 128-col strip with V_WMMA_F32_16X16X4_F32.
// W is cooperatively staged into LDS in WMMA-B-fragment layout (float2 per
// lane, [kstep][tile][lane]), double-buffered in 32-k-row chunks so the inner
// loop is a single conflict-free ds_load_b64 per (tile, kstep).
// RELU (compile-time) applies ReLU to the A operand on load, fusing the
// previous layer's activation.
// ---------------------------------------------------------------------------
template <int K, int RELU>
__global__ __launch_bounds__(256) void gcn_gemm_wmma(
    const float* __restrict__ X, const float* __restrict__ W,
    const float* __restrict__ bias, float* __restrict__ Y, int M) {
  // [buf][kstep(8)][tile(8)][lane(32)] float2  -> 2 * 8*8*32*2 floats = 32 KB
  __shared__ float wlds[2 * 8 * 8 * 32 * 2];

  const int tid   = threadIdx.x;
  const int lane  = tid & 31;
  const int wave  = tid >> 5;
  const int strip = blockIdx.x * 8 + wave;        // wave-uniform
  const bool active = (strip * 16 < M);           // uniform within wave

  const int row0 = strip * 16;
  // A-matrix fp32 16x4 layout: lanes 0-15 hold (K=0,1), lanes 16-31 (K=2,3)
  const int rA   = row0 + (lane & 15);
  const int kOff = (lane >> 4) * 2;               // 0 or 2
  const int colB = lane & 15;

  v8f acc[8];
#pragma unroll
  for (int t = 0; t < 8; ++t) acc[t] = (v8f){0.f,0.f,0.f,0.f,0.f,0.f,0.f,0.f};

  const float* __restrict__ xrow = X + (size_t)(active ? rA : 0) * K;

  const int NCH = K / 32;                         // 32 k-rows (8 wmma steps) per chunk
  for (int ch = 0; ch < NCH; ++ch) {
    // ---- cooperative stage of chunk ch into buffer (ch&1), all threads ----
    float* buf = wlds + (ch & 1) * (8 * 8 * 32 * 2);
#pragma unroll
    for (int q = 0; q < 4; ++q) {
      const int fi = q * 256 + tid;               // float4 index within chunk
      const int kk = fi >> 5;                     // k-row in chunk (0..31)
      const int c0 = (fi & 31) * 4;               // column (0..124)
      const float4 v =
          *(const float4*)(W + (size_t)(ch * 32 + kk) * DOUT + c0);
      const int ks = kk >> 2;                     // wmma k-step in chunk
      const int r  = kk & 3;                      // k within step
      const int ln = (r & 2) << 3;                // lane-half select (0 / 16)
      const int el = r & 1;                       // float2 element
      const float vv[4] = {v.x, v.y, v.z, v.w};
#pragma unroll
      for (int j = 0; j < 4; ++j) {
        const int c    = c0 + j;
        const int tile = c >> 4;
        const int col  = c & 15;
        buf[(((ks * 8 + tile) * 32) + (ln | col)) * 2 + el] = vv[j];
      }
    }
    __syncthreads();  // single barrier/chunk: double buffering covers WAR

    // ---- 8 wmma k-steps from LDS ----
    if (active) {
      const float* __restrict__ xk = xrow + ch * 32;
#pragma unroll
      for (int ks = 0; ks < 8; ++ks) {
        float ax = xk[ks * 4 + kOff];
        float ay = xk[ks * 4 + kOff + 1];
        if (RELU) { ax = fmaxf(ax, 0.f); ay = fmaxf(ay, 0.f); }
        v2f a; a.x = ax; a.y = ay;
        const float* __restrict__ bb = buf + ks * 8 * 32 * 2;
#pragma unroll
        for (int t = 0; t < 8; ++t) {
          const v2f b = *(const v2f*)(bb + (t * 32 + lane) * 2);  // ds_load_b64
          acc[t] = __builtin_amdgcn_wmma_f32_16x16x4_f32(
              false, a, false, b, (short)0, acc[t], false, false);
        }
      }
    }
  }

  if (!active) return;
  // C/D layout: VGPR g -> row (row0+g) for lanes 0-15, (row0+g+8) for 16-31
  const int mh = (lane >> 4) * 8;
#pragma unroll
  for (int t = 0; t < 8; ++t) {
    const int   c  = t * 16 + colB;
    const float bv = bias[c];
#pragma unroll
    for (int g = 0; g < 8; ++g) {
      const int row = row0 + g + mh;
      Y[(size_t)row * DOUT + c] = acc[t][g] + bv;
    }
  }
}

// ---------------------------------------------------------------------------
// Edge aggregation: out[dst[e]] += H[src[e]] * w[e]  (128 floats per edge)
// 32 threads per edge, float4 per thread. Gathers are L2-resident (51.2 MB
// feature matrix << 192 MB L2); scatter uses no-return global_atomic_add_f32.
// ---------------------------------------------------------------------------
__global__ __launch_bounds__(256) void gcn_edge_agg(
    const float* __restrict__ H, const int* __restrict__ src,
    const int* __restrict__ dst, const float* __restrict__ w,
    float* __restrict__ out, int E) {
  const long long total  = (long long)E * 32;
  const long long stride = (long long)gridDim.x * blockDim.x;
  for (long long i = (long long)blockIdx.x * blockDim.x + threadIdx.x;
       i < total; i += stride) {
    const int e    = (int)(i >> 5);
    const int lane = (int)(i & 31);
    const int s    = src[e];
    const int d    = dst[e];
    const float we = w[e];
    const float4 v = *(const float4*)(H + (size_t)s * DOUT + lane * 4);
    float* o = out + (size_t)d * DOUT + lane * 4;
    atomicAdd(o + 0, v.x * we);
    atomicAdd(o + 1, v.y * we);
    atomicAdd(o + 2, v.z * we);
    atomicAdd(o + 3, v.w * we);
  }
}

extern "C" void kernel_launch(void* const* d_in, const int* in_sizes, int n_in,
                              void* d_out, int out_size, void* d_ws, size_t ws_size,
                              hipStream_t stream) {
  const float* x    = (const float*)d_in[0];
  const float* ew   = (const float*)d_in[1];
  const float* W1   = (const float*)d_in[2];
  const float* b1   = (const float*)d_in[3];
  const float* W2   = (const float*)d_in[4];
  const float* b2   = (const float*)d_in[5];
  const float* W3   = (const float*)d_in[6];
  const float* b3   = (const float*)d_in[7];
  const int*   esrc = (const int*)d_in[8];
  const int*   edst = (const int*)d_in[9];
  float* out = (float*)d_out;

  const int IN_DIM = 256;
  const int N = in_sizes[0] / IN_DIM;   // 100000
  const int E = in_sizes[1];            // 1600000

  float* t0 = (float*)d_ws;                       // N x 128 (GEMM output)
  float* a0 = (float*)d_ws + (size_t)N * DOUT;    // N x 128 (aggregation)
  const size_t featBytes = (size_t)N * DOUT * sizeof(float);

  const int strips   = (N + 15) / 16;
  const int gemmGrid = (strips + 7) / 8;          // 8 waves / block
  const int aggGrid  = 8192;

  // ---- Layer 1: agg(x@W1 + b1); relu fused into layer-2 GEMM load
  gcn_gemm_wmma<256, 0><<<gemmGrid, 256, 0, stream>>>(x, W1, b1, t0, N);
  hipMemsetAsync(a0, 0, featBytes, stream);
  gcn_edge_agg<<<aggGrid, 256, 0, stream>>>(t0, esrc, edst, ew, a0, E);

  // ---- Layer 2
  gcn_gemm_wmma<128, 1><<<gemmGrid, 256, 0, stream>>>(a0, W2, b2, t0, N);
  hipMemsetAsync(a0, 0, featBytes, stream);
  gcn_edge_agg<<<aggGrid, 256, 0, stream>>>(t0, esrc, edst, ew, a0, E);

  // ---- Layer 3 (no output relu): scatter straight into d_out
  gcn_gemm_wmma<128, 1><<<gemmGrid, 256, 0, stream>>>(a0, W3, b3, t0, N);
  hipMemsetAsync(out, 0, (size_t)out_size * sizeof(float), stream);
  gcn_edge_agg<<<aggGrid, 256, 0, stream>>>(t0, esrc, edst, ew, out, E);
}